// GraphNetwork_6717328851835
// MI455X (gfx1250) — compile-verified
//
#include <hip/hip_runtime.h>

typedef __attribute__((ext_vector_type(2))) float v2f;
typedef __attribute__((ext_vector_type(8))) float v8f;

#define NN 50000
#define NE 1600000
#define FIN 128
#define FE 16
#define HID 32
#define ETOT (NE + NN)

// ---- monotone float <-> uint mapping for atomic max on floats ----
__device__ __forceinline__ unsigned fmapOrd(float f) {
  unsigned u = __float_as_uint(f);
  return (u & 0x80000000u) ? ~u : (u | 0x80000000u);
}
__device__ __forceinline__ float funmapOrd(unsigned u) {
  return (u & 0x80000000u) ? __uint_as_float(u ^ 0x80000000u) : __uint_as_float(~u);
}

__device__ __forceinline__ float wredsum(float v) {
  v += __shfl_xor(v, 16, 32);
  v += __shfl_xor(v, 8, 32);
  v += __shfl_xor(v, 4, 32);
  v += __shfl_xor(v, 2, 32);
  v += __shfl_xor(v, 1, 32);
  return v;
}

// v1 = We1 @ ae1, v2 = We2 @ ae2 (16-vectors); zero the edge-scalar sums.
__global__ void k_prep(const float* __restrict__ We1, const float* __restrict__ ae1,
                       const float* __restrict__ We2, const float* __restrict__ ae2,
                       float* __restrict__ v1, float* __restrict__ v2,
                       float* __restrict__ sums) {
  int t = threadIdx.x;  // 32 threads
  if (t < 16) {
    float a = 0.f, b = 0.f;
    for (int c = 0; c < HID; ++c) {
      a += We1[t * HID + c] * ae1[c];
      b += We2[t * HID + c] * ae2[c];
    }
    v1[t] = a;
    v2[t] = b;
  }
  if (t < 2) sums[t] = 0.f;
}

// One pass over edge_attr: ec1[e] = ea[e].v1, ec2[e] = ea[e].v2, plus global sums
// (mean(ea).v == sum(ec)/E gives the self-loop contribution for free).
__global__ void k_edge_ec(const float* __restrict__ ea, const float* __restrict__ v1,
                          const float* __restrict__ v2, float* __restrict__ ec1,
                          float* __restrict__ ec2, float* __restrict__ sums) {
  __shared__ float sv[32];
  if (threadIdx.x < 32)
    sv[threadIdx.x] = (threadIdx.x < 16) ? v1[threadIdx.x] : v2[threadIdx.x - 16];
  __syncthreads();
  int e = blockIdx.x * blockDim.x + threadIdx.x;
  float d1 = 0.f, d2 = 0.f;
  if (e < NE) {
    const float4* p = (const float4*)(ea + (size_t)e * FE);
#pragma unroll
    for (int q = 0; q < 4; ++q) {
      float4 f = p[q];
      d1 += f.x * sv[q * 4 + 0] + f.y * sv[q * 4 + 1] + f.z * sv[q * 4 + 2] + f.w * sv[q * 4 + 3];
      d2 += f.x * sv[16 + q * 4 + 0] + f.y * sv[16 + q * 4 + 1] + f.z * sv[16 + q * 4 + 2] +
            f.w * sv[16 + q * 4 + 3];
    }
    ec1[e] = d1;
    ec2[e] = d2;
  }
  float s1 = wredsum(d1);
  float s2 = wredsum(d2);
  if ((threadIdx.x & 31) == 0) {
    atomicAdd(&sums[0], s1);
    atomicAdd(&sums[1], s2);
  }
}

// C[NNxHID] = A[NNxK] @ B[KxHID] via v_wmma_f32_16x16x4_f32.
// One wave computes a 16x32 tile: two 16x16 accumulators, K stepped by 4.
__global__ void k_gemm_wmma(const float* __restrict__ A, const float* __restrict__ B,
                            float* __restrict__ C, int K) {
  int tile = blockIdx.x * (blockDim.x >> 5) + (threadIdx.x >> 5);
  if (tile * 16 >= NN) return;  // wave-uniform: EXEC stays all-1s for WMMA
  int lane = threadIdx.x & 31;
  int half = lane >> 4;  // 0: K sub 0..1 / M 0..7 of C ; 1: K sub 2..3 / M 8..15
  int lm = lane & 15;    // A: M row ; B/C: N column
  int m0 = tile * 16;
  v8f c0 = {};  // N = 0..15
  v8f c1 = {};  // N = 16..31
  const float* arow = A + (size_t)(m0 + lm) * K + 2 * half;
  for (int k = 0; k < K; k += 4) {
    v2f a;
    a.x = arow[k];
    a.y = arow[k + 1];
    const float* bp = B + (size_t)(k + 2 * half) * HID;
    v2f b0, b1;
    b0.x = bp[lm];
    b0.y = bp[HID + lm];
    b1.x = bp[lm + 16];
    b1.y = bp[HID + lm + 16];
    c0 = __builtin_amdgcn_wmma_f32_16x16x4_f32(false, a, false, b0, (short)0, c0, false, false);
    c1 = __builtin_amdgcn_wmma_f32_16x16x4_f32(false, a, false, b1, (short)0, c1, false, false);
  }
#pragma unroll
  for (int r = 0; r < 8; ++r) {
    int m = m0 + r + 8 * half;  // C layout: VGPR r -> M=r (lanes 0-15) / M=8+r (lanes 16-31)
    C[(size_t)m * HID + lm] = c0[r];
    C[(size_t)m * HID + 16 + lm] = c1[r];
  }
}

// Per-node: s_src = h.a_src, s_dst = h.a_dst; init amax/denom/agg. Wave per node.
__global__ void k_node_pre(const float* __restrict__ h, const float* __restrict__ a_src,
                           const float* __restrict__ a_dst, float* __restrict__ ssrc,
                           float* __restrict__ sdst, unsigned* __restrict__ amaxu,
                           float* __restrict__ denom, float* __restrict__ agg) {
  int n = blockIdx.x * (blockDim.x >> 5) + (threadIdx.x >> 5);
  if (n >= NN) return;
  int c = threadIdx.x & 31;
  float hv = h[(size_t)n * HID + c];
  float s1 = wredsum(hv * a_src[c]);
  float s2 = wredsum(hv * a_dst[c]);
  agg[(size_t)n * HID + c] = 0.f;
  if (c == 0) {
    ssrc[n] = s1;
    sdst[n] = s2;
    amaxu[n] = 0u;  // below any mapped finite float; every node has a self-loop
    denom[n] = 0.f;
  }
}

// Per-edge (incl. self-loops): alpha = leaky(s_src[src]+s_dst[dst]+ec), segment max.
__global__ void k_alpha(const int* __restrict__ ei, const float* __restrict__ ssrc,
                        const float* __restrict__ sdst, const float* __restrict__ ec,
                        const float* __restrict__ sums, int sumIdx,
                        float* __restrict__ alpha, unsigned* __restrict__ amaxu) {
  int e = blockIdx.x * blockDim.x + threadIdx.x;
  if (e >= ETOT) return;
  int src, dst;
  float ecv;
  if (e < NE) {
    src = ei[e];
    dst = ei[NE + e];
    ecv = ec[e];
  } else {
    src = dst = e - NE;
    ecv = sums[sumIdx] * (1.0f / NE);  // mean edge_attr projected
  }
  float a = ssrc[src] + sdst[dst] + ecv;
  a = (a >= 0.f) ? a : 0.2f * a;  // leaky_relu
  alpha[e] = a;
  atomicMax(&amaxu[dst], fmapOrd(a));
}

// Wave per edge: e = exp(alpha - amax[dst]); denom[dst]+=e; agg[dst] += e*h[src].
__global__ void k_expagg(const int* __restrict__ ei, const float* __restrict__ alpha,
                         const unsigned* __restrict__ amaxu, float* __restrict__ denom,
                         const float* __restrict__ h, float* __restrict__ agg) {
  int e = blockIdx.x * (blockDim.x >> 5) + (threadIdx.x >> 5);
  if (e >= ETOT) return;
  int c = threadIdx.x & 31;
  int src, dst;
  if (e < NE) {
    src = ei[e];
    dst = ei[NE + e];
  } else {
    src = dst = e - NE;
  }
  float ex = expf(alpha[e] - funmapOrd(amaxu[dst]));
  if (c == 0) atomicAdd(&denom[dst], ex);
  atomicAdd(&agg[(size_t)dst * HID + c], ex * h[(size_t)src * HID + c]);
}

// out = agg/(denom+EPS) + bias, optional relu. In-place on agg.
__global__ void k_node_post(float* __restrict__ agg, const float* __restrict__ denom,
                            const float* __restrict__ bias, int doRelu) {
  int i = blockIdx.x * blockDim.x + threadIdx.x;
  if (i >= NN * HID) return;
  int n = i >> 5, c = i & 31;
  float v = agg[i] / (denom[n] + 1e-16f) + bias[c];
  if (doRelu) v = fmaxf(v, 0.f);
  agg[i] = v;
}

// Wave per original edge: out[e] = concat(h[src],h[dst]) . Ws + bs
__global__ void k_score(const int* __restrict__ ei, const float* __restrict__ h,
                        const float* __restrict__ Ws, const float* __restrict__ bs,
                        float* __restrict__ out) {
  int e = blockIdx.x * (blockDim.x >> 5) + (threadIdx.x >> 5);
  if (e >= NE) return;
  int c = threadIdx.x & 31;
  int s = ei[e], d = ei[NE + e];
  float acc = h[(size_t)s * HID + c] * Ws[c] + h[(size_t)d * HID + c] * Ws[HID + c];
  acc = wredsum(acc);
  if (c == 0) out[e] = acc + bs[0];
}

extern "C" void kernel_launch(void* const* d_in, const int* in_sizes, int n_in,
                              void* d_out, int out_size, void* d_ws, size_t ws_size,
                              hipStream_t stream) {
  const float* x = (const float*)d_in[0];
  const int* ei = (const int*)d_in[1];
  const float* ea = (const float*)d_in[2];
  const float* W1 = (const float*)d_in[3];
  const float* as1 = (const float*)d_in[4];
  const float* ad1 = (const float*)d_in[5];
  const float* We1 = (const float*)d_in[6];
  const float* ae1 = (const float*)d_in[7];
  const float* b1 = (const float*)d_in[8];
  const float* W2 = (const float*)d_in[9];
  const float* as2 = (const float*)d_in[10];
  const float* ad2 = (const float*)d_in[11];
  const float* We2 = (const float*)d_in[12];
  const float* ae2 = (const float*)d_in[13];
  const float* b2 = (const float*)d_in[14];
  const float* Ws = (const float*)d_in[15];
  const float* bs = (const float*)d_in[16];
  float* out = (float*)d_out;

  float* w = (float*)d_ws;
  float* hA = w;          w += (size_t)NN * HID;   // h1, then h2-pre
  float* agg = w;         w += (size_t)NN * HID;   // accum, then layer output (in-place)
  float* alpha = w;       w += ETOT;
  float* ec1 = w;         w += NE;
  float* ec2 = w;         w += NE;
  float* ssrc = w;        w += NN;
  float* sdst = w;        w += NN;
  unsigned* amaxu = (unsigned*)w; w += NN;
  float* denom = w;       w += NN;
  float* v1 = w;          w += 16;
  float* v2 = w;          w += 16;
  float* sums = w;        w += 2;

  dim3 b256(256);
  k_prep<<<1, 32, 0, stream>>>(We1, ae1, We2, ae2, v1, v2, sums);
  k_edge_ec<<<NE / 256, b256, 0, stream>>>(ea, v1, v2, ec1, ec2, sums);

  // ---- layer 1 ----
  k_gemm_wmma<<<(3125 + 7) / 8, b256, 0, stream>>>(x, W1, hA, FIN);
  k_node_pre<<<NN / 8, b256, 0, stream>>>(hA, as1, ad1, ssrc, sdst, amaxu, denom, agg);
  k_alpha<<<(ETOT + 255) / 256, b256, 0, stream>>>(ei, ssrc, sdst, ec1, sums, 0, alpha, amaxu);
  k_expagg<<<ETOT / 8, b256, 0, stream>>>(ei, alpha, amaxu, denom, hA, agg);
  k_node_post<<<(NN * HID) / 256, b256, 0, stream>>>(agg, denom, b1, 1);

  // ---- layer 2 (agg now holds relu(h1) input) ----
  k_gemm_wmma<<<(3125 + 7) / 8, b256, 0, stream>>>(agg, W2, hA, HID);
  k_node_pre<<<NN / 8, b256, 0, stream>>>(hA, as2, ad2, ssrc, sdst, amaxu, denom, agg);
  k_alpha<<<(ETOT + 255) / 256, b256, 0, stream>>>(ei, ssrc, sdst, ec2, sums, 1, alpha, amaxu);
  k_expagg<<<ETOT / 8, b256, 0, stream>>>(ei, alpha, amaxu, denom, hA, agg);
  k_node_post<<<(NN * HID) / 256, b256, 0, stream>>>(agg, denom, b2, 0);

  // ---- edge scorer over original edges (agg holds final node embeddings) ----
  k_score<<<NE / 8, b256, 0, stream>>>(ei, agg, Ws, bs, out);
}